// seq2seq_18511309045981
// MI455X (gfx1250) — compile-verified
//
#include <hip/hip_runtime.h>

// ---------------------------------------------------------------------------
// CDNA5 (gfx1250) fp32 GRU seq2seq using V_WMMA_F32_16X16X4_F32.
// B=64, T=100, D=75, H=1024 (encoder per-dir), H=2048 (decoder).
//
// Each GRU pass is ONE persistent kernel: 128 blocks loop over T on-device,
// synchronizing with a grid-wide atomic barrier. A block = 8 waves =
// 4 column tiles x 2 K-slices: wave pairs split the contraction dimension
// and combine partial accumulators through a padded LDS buffer. Wave-uniform
// values (wave id, K-slice bounds) are pinned to SGPRs via readfirstlane so
// the k-loops have scalar trip counts and unroll into load->wmma streams.
// ---------------------------------------------------------------------------

typedef __attribute__((ext_vector_type(2))) float v2f;
typedef __attribute__((ext_vector_type(8))) float v8f;

#define WMMA_F32(a, b, c) \
  __builtin_amdgcn_wmma_f32_16x16x4_f32(false, (a), false, (b), (short)0, (c), false, false)

__device__ __forceinline__ v2f ldfrag(const float* __restrict__ p) {
  v2f f;
  f.x = p[0];
  f.y = p[1];
  return f;
}

// Branchless guarded load for K-tails: clamp index (always-valid address),
// load unconditionally, zero via select (v_cndmask, no EXEC manipulation).
__device__ __forceinline__ v2f ldfrag_tail(const float* __restrict__ rowp,
                                           int kb, int K) {
  const int i0 = (kb < K) ? kb : 0;
  const int i1 = (kb + 1 < K) ? kb + 1 : 0;
  const float a0 = rowp[i0];
  const float a1 = rowp[i1];
  v2f f;
  f.x = (kb < K) ? a0 : 0.0f;
  f.y = (kb + 1 < K) ? a1 : 0.0f;
  return f;
}

__device__ __forceinline__ float sigmoidf_(float x) {
  return 1.0f / (1.0f + expf(-x));
}

// Grid-wide barrier: arrive counter bar[0], generation bar[1].
// All blocks must be co-resident (128 blocks <= #WGPs on MI455X).
__device__ __forceinline__ void grid_sync(unsigned* bar, unsigned nblocks) {
  __threadfence();                 // make this thread's y-stores visible
  __syncthreads();                 // all LDS reads of this round done
  if (threadIdx.x == 0) {
    volatile unsigned* vb = (volatile unsigned*)bar;
    const unsigned gen = vb[1];    // safe: gen can't advance until we arrive
    const unsigned prev = atomicAdd(&bar[0], 1u);
    if (prev == nblocks - 1u) {
      vb[0] = 0u;                  // reset for next round
      __threadfence();
      atomicAdd(&bar[1], 1u);      // release new generation
    } else {
      while (vb[1] == gen) { __builtin_amdgcn_s_sleep(1); }
    }
    __threadfence();
  }
  __syncthreads();
}

__global__ void init_bar(unsigned* bar) {
  if (threadIdx.x < 2) bar[threadIdx.x] = 0u;
}

// Persistent GRU pass over T timesteps (see header comment).
//   gi = x_t @ Wih^T       (skipped when Kin == 0, e.g. decoder)
//   gh = h_{t-1} @ Whh^T   (zero at t==0 when h0_zero: LDS staged with zeros)
//   r = sig(gi_r+gh_r+b);  z = sig(gi_z+gh_z+b)
//   n = tanh(gi_n+bi_n + r*(gh_n+bh_n));  h_new = (1-z)*n + z*h_prev
// Grid: exactly (H/16)*4*numDirs*2 waves = 128 blocks of 256 threads.
// Dynamic LDS: 16*(Kin+H)*4 bytes staging + 4*32*33*4 bytes reduction.
__global__ void gru_seq_wmma(
    const float* __restrict__ xin, long x_t_stride, long x_row_stride, int Kin,
    const float* __restrict__ Wih, long Wih_dir_stride,   // rows: 3H x Kin
    const float* __restrict__ Whh, long Whh_dir_stride,   // rows: 3H x H
    const float* __restrict__ bias, long bias_dir_stride, // [bih(3H)|bhh(3H)]
    const float* __restrict__ h0, long h0_row_stride, int h0_zero,
    float* __restrict__ y, long y_t_stride, long y_row_stride,
    int T, int H, unsigned* bar, unsigned nblocks) {
  extern __shared__ float sh[];
  const int tid  = threadIdx.x;
  const int lane = tid & 31;
  // Wave-uniform values pinned to SGPRs -> scalar loop bounds / branches.
  const int wvu = __builtin_amdgcn_readfirstlane(tid >> 5);
  const int ctl = wvu & 3;                   // column tile within block
  const int ks  = wvu >> 2;                  // K-slice (0 = low, 1 = high)

  const int colTiles = H >> 4;               // 64 or 128 (multiple of 4)
  const int bpr = colTiles >> 2;             // blocks per (rt, dir) row
  const int ct  = ((blockIdx.x % bpr) << 2) + ctl;
  const int rt  = (blockIdx.x / bpr) & 3;    // block-uniform row tile
  const int dir = blockIdx.x / (bpr * 4);    // block-uniform direction

  const int m0 = rt << 4;
  const int ln = lane & 15;
  const int kh = lane >> 4;
  const int kadd = kh << 1;
  const int bcol = (ct << 4) + ln;           // gate column in [0, H)

  float* shx  = sh;                          // [16, Kin] x_t rows
  float* shh  = sh + 16 * Kin;               // [16, H]   h_{t-1} rows
  float* shred = shh + 16 * H;               // [4][32 lanes][33] partials

  const float* WdI = Wih + (long)dir * Wih_dir_stride;
  const float* WdH = Whh + (long)dir * Whh_dir_stride;

  // K-slice bounds (all scalar): input dim may be odd (75 -> 40/35).
  const int Kin0  = ((Kin + 7) >> 3) << 2;   // low-slice length (mult of 4)
  const int kbegI = ks ? Kin0 : 0;
  const int kendI = ks ? Kin : Kin0;
  const int mainI = (kendI - kbegI) & ~3;
  const int kbegH = ks * (H >> 1);
  const int Hh = H >> 1;

  const float* BIr = WdI + (long)(bcol)         * Kin + kbegI + kadd;
  const float* BIz = WdI + (long)(H + bcol)     * Kin + kbegI + kadd;
  const float* BIn = WdI + (long)(2 * H + bcol) * Kin + kbegI + kadd;
  const float* BHr = WdH + (long)(bcol)         * H + kbegH + kadd;
  const float* BHz = WdH + (long)(H + bcol)     * H + kbegH + kadd;
  const float* BHn = WdH + (long)(2 * H + bcol) * H + kbegH + kadd;

  const float* bih = bias + (long)dir * bias_dir_stride;
  const float* bhh = bih + 3 * H;
  const float b_r  = bih[bcol]         + bhh[bcol];
  const float b_z  = bih[H + bcol]     + bhh[H + bcol];
  const float bi_n = bih[2 * H + bcol];
  const float bh_n = bhh[2 * H + bcol];

  const int H4 = H >> 2;                     // float4 chunks per h row
  const bool x_vec4 = (Kin & 127) == 0;      // float4-friendly x staging
  const int Kin4 = Kin >> 2;

  for (int t = 0; t < T; ++t) {
    // ---- stage x_t rows (block's 16 batch rows) into LDS ----------------
    if (Kin > 0) {
      const float* xs = xin + (long)t * x_t_stride;
      for (int r = wvu; r < 16; r += 8) {
        const float* src = xs + (long)(m0 + r) * x_row_stride;
        float* dst = shx + r * Kin;
        if (x_vec4) {                        // b128 path (Kin % 128 == 0)
          const float4* s4 = (const float4*)src;
          float4* d4 = (float4*)dst;
          for (int c0 = 0; c0 < Kin4; c0 += 32) d4[c0 + lane] = s4[c0 + lane];
        } else {                             // scalar path (Kin = 75)
          int c0 = 0;
          for (; c0 + 32 <= Kin; c0 += 32) dst[c0 + lane] = src[c0 + lane];
          if (c0 + lane < Kin) dst[c0 + lane] = src[c0 + lane];
        }
      }
    }
    // ---- stage h_{t-1} rows into LDS (zeros at t==0 when h0_zero) -------
    const bool h_is_zero = (t == 0) && (h0_zero != 0);
    if (h_is_zero) {
      const float4 z4 = {0.f, 0.f, 0.f, 0.f};
      for (int r = wvu; r < 16; r += 8) {
        float4* d4 = (float4*)(shh + r * H);
        for (int c0 = 0; c0 < H4; c0 += 32) d4[c0 + lane] = z4;
      }
    } else {
      const float* hsrc = (t == 0) ? h0 : (y + (long)(t - 1) * y_t_stride);
      const long hrs    = (t == 0) ? h0_row_stride : y_row_stride;
      for (int r = wvu; r < 16; r += 8) {
        const float4* s4 =
            (const float4*)(hsrc + (long)(m0 + r) * hrs + (long)dir * H);
        float4* d4 = (float4*)(shh + r * H);
        for (int c0 = 0; c0 < H4; c0 += 32) d4[c0 + lane] = s4[c0 + lane];
      }
    }
    __syncthreads();

    v8f acc_r  = {0.f, 0.f, 0.f, 0.f, 0.f, 0.f, 0.f, 0.f};
    v8f acc_z  = {0.f, 0.f, 0.f, 0.f, 0.f, 0.f, 0.f, 0.f};
    v8f acc_in = {0.f, 0.f, 0.f, 0.f, 0.f, 0.f, 0.f, 0.f};
    v8f acc_hn = {0.f, 0.f, 0.f, 0.f, 0.f, 0.f, 0.f, 0.f};

    // ---- input-gate GEMM (this wave's K-slice): A from LDS, B global ----
    if (Kin > 0) {
      const float* pa = shx + ln * Kin + kbegI + kadd;
      const float* pr = BIr;
      const float* pz = BIz;
      const float* pn = BIn;
#pragma unroll 4
      for (int k0 = 0; k0 < mainI; k0 += 4) {   // scalar trip count
        v2f a  = ldfrag(pa);
        v2f br = ldfrag(pr);
        v2f bz = ldfrag(pz);
        v2f bn = ldfrag(pn);
        acc_r  = WMMA_F32(a, br, acc_r);
        acc_z  = WMMA_F32(a, bz, acc_z);
        acc_in = WMMA_F32(a, bn, acc_in);
        pa += 4; pr += 4; pz += 4; pn += 4;
      }
      if (kbegI + mainI < kendI) {           // branchless tail (Kin = 75)
        const int kb = kbegI + mainI + kadd;
        v2f a  = ldfrag_tail(shx + ln * Kin, kb, kendI);
        v2f br = ldfrag_tail(WdI + (long)(bcol)         * Kin, kb, kendI);
        v2f bz = ldfrag_tail(WdI + (long)(H + bcol)     * Kin, kb, kendI);
        v2f bn = ldfrag_tail(WdI + (long)(2 * H + bcol) * Kin, kb, kendI);
        acc_r  = WMMA_F32(a, br, acc_r);
        acc_z  = WMMA_F32(a, bz, acc_z);
        acc_in = WMMA_F32(a, bn, acc_in);
      }
    }

    // ---- hidden-gate GEMM (this wave's K-slice) -------------------------
    if (!h_is_zero) {                        // block-uniform skip at t==0
      const float* pa = shh + ln * H + kbegH + kadd;
      const float* pr = BHr;
      const float* pz = BHz;
      const float* pn = BHn;
#pragma unroll 4
      for (int k0 = 0; k0 < Hh; k0 += 4) {   // scalar trip count
        v2f a  = ldfrag(pa);
        v2f br = ldfrag(pr);
        v2f bz = ldfrag(pz);
        v2f bn = ldfrag(pn);
        acc_r  = WMMA_F32(a, br, acc_r);
        acc_z  = WMMA_F32(a, bz, acc_z);
        acc_hn = WMMA_F32(a, bn, acc_hn);
        pa += 4; pr += 4; pz += 4; pn += 4;
      }
    }

    // ---- combine K-slices through LDS (stride 33: bank-conflict-free) ---
    float* red = shred + (ctl * 32 + lane) * 33;
    if (ks == 1) {                           // scalar branch
#pragma unroll
      for (int g = 0; g < 8; ++g) {
        red[g]      = acc_r[g];
        red[8 + g]  = acc_z[g];
        red[16 + g] = acc_in[g];
        red[24 + g] = acc_hn[g];
      }
    }
    __syncthreads();

    if (ks == 0) {                           // scalar branch
#pragma unroll
      for (int g = 0; g < 8; ++g) {
        acc_r[g]  += red[g];
        acc_z[g]  += red[8 + g];
        acc_in[g] += red[16 + g];
        acc_hn[g] += red[24 + g];
      }
      // ---- epilogue: gates + state update; h_prev from LDS --------------
      // C/D layout: vgpr g, lanes 0-15 -> M = g ; lanes 16-31 -> M = g + 8.
#pragma unroll
      for (int g = 0; g < 8; ++g) {
        const int rloc = g + (kh << 3);
        const float hp = shh[rloc * H + bcol];         // 0 when h_is_zero
        const float r = sigmoidf_(acc_r[g] + b_r);
        const float z = sigmoidf_(acc_z[g] + b_z);
        const float n = tanhf(acc_in[g] + bi_n + r * (acc_hn[g] + bh_n));
        const float hnew = (1.0f - z) * n + z * hp;
        y[(long)t * y_t_stride + (long)(m0 + rloc) * y_row_stride +
          (long)dir * H + bcol] = hnew;
      }
    }

    grid_sync(bar, nblocks);                 // y[t] visible before t+1
  }
}

// enc_hidden[b, j] = y[b, clamp(seq_len[b],1,T)-1, j]   (y: [B,T,HH])
__global__ void gather_last(const float* __restrict__ y,
                            const int* __restrict__ seq,
                            float* __restrict__ out, int T, int HH) {
  const int idx = blockIdx.x * blockDim.x + threadIdx.x;
  const int b = idx / HH;
  const int j = idx - b * HH;
  int sl = seq[b];
  sl = sl < 1 ? 1 : (sl > T ? T : sl);
  out[idx] = y[((long)b * T + (sl - 1)) * (long)HH + j];
}

// dec_out[m, n] = ys[m, :] @ W[n, :] + bias[n]; N = 75 -> column-guarded.
__global__ void proj_wmma(const float* __restrict__ ys,
                          const float* __restrict__ W,
                          const float* __restrict__ bias,
                          float* __restrict__ out, int M, int N, int K) {
  const int lane = threadIdx.x & 31;
  const int wave = blockIdx.x * (blockDim.x >> 5) +
                   __builtin_amdgcn_readfirstlane(threadIdx.x >> 5);
  const int colTiles = (N + 15) >> 4;
  const int ct = wave % colTiles;
  const int rt = wave / colTiles;
  const int ln = lane & 15;
  const int kh = lane >> 4;
  const int kadd = kh << 1;
  const int m = rt * 16 + ln;
  const int n = ct * 16 + ln;
  const bool nok = (n < N);
  const float nmask = nok ? 1.0f : 0.0f;

  v8f acc = {0.f, 0.f, 0.f, 0.f, 0.f, 0.f, 0.f, 0.f};
  const float* pa = ys + (long)m * K + kadd;
  const float* pb = W + (long)(nok ? n : 0) * K + kadd;  // clamped: valid
#pragma unroll 4
  for (int k0 = 0; k0 < K; k0 += 4) {
    v2f a = ldfrag(pa);
    v2f braw = ldfrag(pb);
    v2f b;
    b.x = nmask * braw.x;
    b.y = nmask * braw.y;
    acc = WMMA_F32(a, b, acc);
    pa += 4; pb += 4;
  }
  const float bb = nok ? bias[n] : 0.0f;
#pragma unroll
  for (int g = 0; g < 8; ++g) {
    const int row = rt * 16 + g + (kh << 3);
    const int col = ct * 16 + ln;
    if (col < N) out[(long)row * N + col] = acc[g] + bb;
  }
}

extern "C" void kernel_launch(void* const* d_in, const int* in_sizes, int n_in,
                              void* d_out, int out_size, void* d_ws, size_t ws_size,
                              hipStream_t stream) {
  (void)in_sizes; (void)n_in; (void)out_size; (void)ws_size;

  const float* x     = (const float*)d_in[0];   // [64,100,75]
  const int*   seq   = (const int*)  d_in[1];   // [64]
  const float* eWih0 = (const float*)d_in[2];   // [2,3072,75]
  const float* eWhh0 = (const float*)d_in[3];   // [2,3072,1024]
  const float* eb0   = (const float*)d_in[4];   // [2,2,3072]
  const float* eWih  = (const float*)d_in[5];   // [2,2,3072,2048]
  const float* eWhh  = (const float*)d_in[6];   // [2,2,3072,1024]
  const float* eb    = (const float*)d_in[7];   // [2,2,2,3072]
  const float* dWih  = (const float*)d_in[8];   // [6144,75]  (input is zeros)
  const float* dWhh  = (const float*)d_in[9];   // [6144,2048]
  const float* db    = (const float*)d_in[10];  // [2,6144]
  const float* oW    = (const float*)d_in[11];  // [75,2048]
  const float* ob    = (const float*)d_in[12];  // [75]
  float* out = (float*)d_out;  // enc_hidden(64*2048) | dec_out(64*100*75)

  const int B = 64, T = 100, D = 75, H = 1024, H2 = 2048;
  const long yElems = (long)B * T * H2;        // 13,107,200 floats / buffer
  float* yA = (float*)d_ws;
  float* yB = yA + yElems;
  unsigned* bar = (unsigned*)(yB + yElems);

  const unsigned NB = 128;                     // blocks per persistent pass
  const size_t redB = (size_t)4 * 32 * 33 * 4;             // 16,896 B
  const size_t lds0   = (size_t)16 * (D + H) * 4 + redB;   // ~85 KB
  const size_t lds12  = (size_t)16 * (H2 + H) * 4 + redB;  // ~209 KB
  const size_t ldsDec = (size_t)16 * (0 + H2) * 4 + redB;  // ~145 KB

  init_bar<<<1, 64, 0, stream>>>(bar);

  // ---- encoder layer 0 (Din = 75), both dirs, persistent, out -> yA -----
  gru_seq_wmma<<<NB, 256, lds0, stream>>>(
      x, (long)D, (long)T * D, D,
      eWih0, (long)3 * H * D,
      eWhh0, (long)3 * H * H,
      eb0, (long)6 * H,
      yA, (long)T * H2, 1,                     // h0 unused (zeros staged)
      yA, (long)H2, (long)T * H2,
      T, H, bar, NB);

  // ---- encoder layers 1..2 (Din = 2H), ping-pong yA <-> yB --------------
  for (int l = 0; l < 2; ++l) {
    const float* lin = (l == 0) ? yA : yB;
    float* lout      = (l == 0) ? yB : yA;
    gru_seq_wmma<<<NB, 256, lds12, stream>>>(
        lin, (long)H2, (long)T * H2, H2,
        eWih + (long)l * 2 * 3 * H * (2 * H), (long)3 * H * (2 * H),
        eWhh + (long)l * 2 * 3 * H * H,       (long)3 * H * H,
        eb   + (long)l * 2 * 2 * 3 * H,       (long)6 * H,
        lout, (long)T * H2, 1,                 // h0 unused (zeros staged)
        lout, (long)H2, (long)T * H2,
        T, H, bar, NB);
  }

  // ---- enc_hidden = yA at last valid step (into d_out[0 .. B*2H)) -------
  gather_last<<<(B * H2) / 256, 256, 0, stream>>>(yA, seq, out, T, H2);

  // ---- decoder: zero input (gi = bih), hidden 2H, h0 = enc_hidden -------
  gru_seq_wmma<<<NB, 256, ldsDec, stream>>>(
      nullptr, 0, 0, 0,                        // Kin = 0: no input GEMM
      dWih, 0,
      dWhh, 0,
      db, 0,
      out, (long)H2, 0,                        // h0 = enc_hidden, nonzero
      yB, (long)H2, (long)T * H2,
      T, H2, bar, NB);

  // ---- dec_out = ys @ out_W^T + out_b  (M=6400, N=75, K=2048) -----------
  const int M = B * T, N = D, K = H2;
  const int projBlocks = (M / 16) * ((N + 15) / 16) / 8;  // 250
  proj_wmma<<<projBlocks, 256, 0, stream>>>(yB, oW, ob, out + (long)B * H2, M, N, K);
}